// CNN_3496103379215
// MI455X (gfx1250) — compile-verified
//
#include <hip/hip_runtime.h>

typedef __attribute__((ext_vector_type(2))) float v2f;
typedef __attribute__((ext_vector_type(8))) float v8f;

#define IMH 1024
#define IMW 1024
#define SROW 24              // staging row stride: 20 halo cols padded to 24
#define NTILES (16*64*64)    // batch * (1024/16)^2

__global__ __launch_bounds__(256) void nconv_pool_wmma(
    const float* __restrict__ data,
    const float* __restrict__ conf,
    const float* __restrict__ weight,   // [2][1][5][5] -> 50 floats
    const float* __restrict__ bias,     // [2]
    float* __restrict__ out_x,          // x1_ds [16][2][512][512]
    float* __restrict__ out_c)          // c1_ds [16][2][512][512]
{
    // Per-wave halo tiles: [conf, data*conf], 20 rows x 24-stride
    __shared__ float lds_tile[8][2][20 * SROW];
    // Precomputed B fragments in per-lane WMMA layout:
    // [(dy*5+ks)*2 + c][lane*2 + comp]  (tile-invariant, built once per block)
    __shared__ float lds_bf[50][64];

    const int tid  = threadIdx.x;
    const int lane = tid & 31;
    const int wv   = tid >> 5;

    // ---- build the banded-Toeplitz B fragment table (once per block) ----
    // B[k][n] = (0 <= k-n <= 4) ? w[c][dy][k-n] : 0, fragment comp -> K = ks*4 + khalf*2 + comp
    for (int idx = tid; idx < 3200; idx += 256) {
        const int laneq = idx & 63;          // lane*2 + comp
        const int entry = idx >> 6;          // (dy*5+ks)*2 + c
        const int comp  = laneq & 1;
        const int ln    = laneq >> 1;
        const int c     = entry & 1;
        const int dyks  = entry >> 1;
        const int dy    = dyks / 5;
        const int ks    = dyks - dy * 5;
        const int nn    = ln & 15;
        const int kh    = ln >> 4;
        const int k0    = ks * 4 + kh * 2 + comp;
        const int d     = k0 - nn;
        const int di    = (d < 0) ? 0 : ((d > 4) ? 4 : d);   // clamped, unconditional load
        const float wv_ = weight[c * 25 + dy * 5 + di];
        lds_bf[entry][laneq] = ((d >= 0) & (d < 5)) ? wv_ : 0.0f;
    }

    // ---- tile decode: one 16x16 conv-output tile per wave ----
    const int t   = blockIdx.x * 8 + wv;
    const int b   = t >> 12;        // / 4096
    const int rem = t & 4095;
    const int ty  = rem >> 6;
    const int tx  = rem & 63;
    const int y0  = ty * 16;
    const int x0  = tx * 16;

    float* sc = &lds_tile[wv][0][0]; // conf halo 20x20 (stride 24)
    float* sn = &lds_tile[wv][1][0]; // data*conf halo

    // ---- stage 20x20 halo (zero-padded at borders), each HBM byte read once ----
    const size_t ibase = (size_t)b * ((size_t)IMH * IMW);
    for (int r = 0; r < 20; ++r) {
        if (lane < 20) {
            int y = y0 - 2 + r;
            int x = x0 - 2 + lane;
            bool ok = (y >= 0) & (y < IMH) & (x >= 0) & (x < IMW);
            size_t gi = ibase + (size_t)(ok ? y : 0) * IMW + (size_t)(ok ? x : 0);
            float cf = ok ? conf[gi] : 0.0f;
            float dt = ok ? data[gi] : 0.0f;
            sc[r * SROW + lane] = cf;
            sn[r * SROW + lane] = dt * cf;
        }
    }
    __syncthreads();

    // per-out-channel kernel sums (wsum) and bias (uniform scalar loads)
    float ws0 = 0.f, ws1 = 0.f;
    #pragma unroll
    for (int i = 0; i < 25; ++i) { ws0 += weight[i]; ws1 += weight[25 + i]; }
    const float inv0 = 1.0f / (ws0 + 1e-20f);
    const float inv1 = 1.0f / (ws1 + 1e-20f);
    const float bi0 = bias[0], bi1 = bias[1];

    const int m     = lane & 15;   // A row (M) for this lane
    const int khalf = lane >> 4;   // which K half this lane group carries

    v8f accD0 = {}, accD1 = {}, accN0 = {}, accN1 = {};

    // ---- banded-Toeplitz conv: D = sum_dy A_dy x B_dy, K=20 in 5 steps of 4 ----
    // Inner loop: 2x ds_load_b64 (A) + 2x ds_load_b64 (B table) + 4x v_wmma.
    #pragma unroll
    for (int dy = 0; dy < 5; ++dy) {
        #pragma unroll
        for (int ks = 0; ks < 5; ++ks) {
            const int k0 = ks * 4 + khalf * 2;
            v2f aC = *(const v2f*)&sc[(m + dy) * SROW + k0];
            v2f aN = *(const v2f*)&sn[(m + dy) * SROW + k0];
            v2f b0 = *(const v2f*)&lds_bf[(dy * 5 + ks) * 2 + 0][lane * 2];
            v2f b1 = *(const v2f*)&lds_bf[(dy * 5 + ks) * 2 + 1][lane * 2];

            accD0 = __builtin_amdgcn_wmma_f32_16x16x4_f32(false, aC, false, b0, (short)0, accD0, false, false);
            accN0 = __builtin_amdgcn_wmma_f32_16x16x4_f32(false, aN, false, b0, (short)0, accN0, false, false);
            accD1 = __builtin_amdgcn_wmma_f32_16x16x4_f32(false, aC, false, b1, (short)0, accD1, false, false);
            accN1 = __builtin_amdgcn_wmma_f32_16x16x4_f32(false, aN, false, b1, (short)0, accN1, false, false);
        }
    }

    // ---- fused epilogue: nconv/cout + 2x2 maxpool with argmax gather ----
    const int hb0 = ty * 8, wb0 = tx * 8;

    #pragma unroll
    for (int c = 0; c < 2; ++c) {
        const float invw = (c == 0) ? inv0 : inv1;
        const float bia  = (c == 0) ? bi0  : bi1;
        // this channel's 16x16 cout -> sc, nconv -> sn (halo no longer needed)
        #pragma unroll
        for (int v = 0; v < 8; ++v) {
            const int mm = v + khalf * 8;          // C/D layout: VGPR v -> M = v + 8*half
            float den = (c == 0) ? accD0[v] : accD1[v];
            float nom = (c == 0) ? accN0[v] : accN1[v];
            float nc  = nom / (den + 1e-20f) + bia;
            sc[mm * 16 + m] = den * invw;          // cout   (N = lane&15 = m)
            sn[mm * 16 + m] = nc;                  // nconv
        }
        __syncthreads();

        #pragma unroll
        for (int i = 0; i < 2; ++i) {
            const int p  = lane + 32 * i;          // 0..63 pooled pixels (8x8)
            const int hb = p >> 3, wb = p & 7;
            const int rr = hb * 2, cc = wb * 2;
            float c00 = sc[rr * 16 + cc],           x00 = sn[rr * 16 + cc];
            float c01 = sc[rr * 16 + cc + 1],       x01 = sn[rr * 16 + cc + 1];
            float c10 = sc[(rr + 1) * 16 + cc],     x10 = sn[(rr + 1) * 16 + cc];
            float c11 = sc[(rr + 1) * 16 + cc + 1], x11 = sn[(rr + 1) * 16 + cc + 1];
            // first-max tie-break in row-major window order (matches argmax)
            float best = c00, xv = x00;
            if (c01 > best) { best = c01; xv = x01; }
            if (c10 > best) { best = c10; xv = x10; }
            if (c11 > best) { best = c11; xv = x11; }
            size_t oi = (((size_t)(b * 2 + c) * 512) + (size_t)(hb0 + hb)) * 512
                        + (size_t)(wb0 + wb);
            out_c[oi] = best * 0.25f;
            out_x[oi] = xv;
        }
        __syncthreads();
    }
}

extern "C" void kernel_launch(void* const* d_in, const int* in_sizes, int n_in,
                              void* d_out, int out_size, void* d_ws, size_t ws_size,
                              hipStream_t stream) {
    const float* data   = (const float*)d_in[0];
    const float* conf   = (const float*)d_in[1];
    const float* weight = (const float*)d_in[2];
    const float* bias   = (const float*)d_in[3];
    float* out_x = (float*)d_out;                       // x1_ds first
    float* out_c = out_x + (size_t)16 * 2 * 512 * 512;  // then c1_ds

    const int blocks = NTILES / 8;                      // 8 wave32 tiles per block
    nconv_pool_wmma<<<blocks, 256, 0, stream>>>(data, conf, weight, bias, out_x, out_c);
}